// PSRoIAlign_38319698215206
// MI455X (gfx1250) — compile-verified
//
#include <hip/hip_runtime.h>

typedef __attribute__((ext_vector_type(2))) float v2f;
typedef __attribute__((ext_vector_type(8))) float v8f;

#define CO     16
#define PHH    7
#define PWW    7
#define NBINS  49           // PH*PW
#define NSAMP  196          // NBINS * SR * SR
#define HH     80
#define WW     80
#define CHW    (HH * WW)    // 6400
#define C_TOT  (CO * NBINS) // 784

__global__ __launch_bounds__(256)
void psroi_align_wmma_kernel(const float* __restrict__ feat,
                             const float* __restrict__ rois,
                             float* __restrict__ out) {
    __shared__ float Vs[CO * NSAMP];   // 12,544 bytes: V[co][s]

    const int n = blockIdx.x;

    // ---- roi parameters (cheap; recomputed per thread) ----
    const float* r = rois + n * 5;
    const int   b  = (int)r[0];
    const float x1 = r[1] * (float)WW;
    const float y1 = r[2] * (float)HH;
    const float x2 = r[3] * (float)WW;
    const float y2 = r[4] * (float)HH;
    const float roi_h = fmaxf(y2 - y1, 0.1f);
    const float roi_w = fmaxf(x2 - x1, 0.1f);
    const float bin_h = roi_h / (float)PHH;
    const float bin_w = roi_w / (float)PWW;

    // ---- Phase 1: gather bilinear samples -> V[16][196] in LDS ----
    for (int idx = threadIdx.x; idx < CO * NSAMP; idx += blockDim.x) {
        const int co  = idx / NSAMP;
        const int s   = idx - co * NSAMP;
        const int bin = s >> 2;               // s = ((ph*7+pw)*2+sy)*2+sx
        const int sub = s & 3;
        const int sy  = sub >> 1;
        const int sx  = sub & 1;
        const int ph  = bin / PWW;
        const int pw  = bin - ph * PWW;

        const float ysv = y1 + (float)ph * bin_h + ((float)sy + 0.5f) * bin_h * 0.5f;
        const float xsv = x1 + (float)pw * bin_w + ((float)sx + 0.5f) * bin_w * 0.5f;

        const bool valid = (ysv >= -1.0f) && (ysv <= (float)HH) &&
                           (xsv >= -1.0f) && (xsv <= (float)WW);

        const float yc = fminf(fmaxf(ysv, 0.0f), (float)(HH - 1));
        const float xc = fminf(fmaxf(xsv, 0.0f), (float)(WW - 1));
        const int y0  = (int)yc;              // yc >= 0, trunc == floor
        const int x0  = (int)xc;
        const int y1i = min(y0 + 1, HH - 1);
        const int x1i = min(x0 + 1, WW - 1);
        const float ly = yc - (float)y0, lx = xc - (float)x0;
        const float hy = 1.0f - ly,      hx = 1.0f - lx;

        const float* fc = feat + (size_t)(b * C_TOT + co * NBINS + bin) * CHW;
        const float v00 = fc[y0 * WW + x0];
        const float v01 = fc[y0 * WW + x1i];
        const float v10 = fc[y1i * WW + x0];
        const float v11 = fc[y1i * WW + x1i];

        const float val = hy * hx * v00 + hy * lx * v01 +
                          ly * hx * v10 + ly * lx * v11;
        Vs[co * NSAMP + s] = valid ? val : 0.0f;
    }
    __syncthreads();

    // ---- Phase 2: out[16 x 49] = V[16 x 196] * P[196 x 49] via WMMA f32 16x16x4 ----
    // Wave id forced scalar so loop control is SALU (EXEC stays all-ones for WMMA).
    const int wave = __builtin_amdgcn_readfirstlane(threadIdx.x >> 5);
    if (wave < 4) {
        const int lane = threadIdx.x & 31;
        const int half = lane >> 4;            // selects K pair for A, M-half for D
        const int m    = lane & 15;            // A row / B,D column
        const int kk0  = wave << 4;

        v8f acc = {0.f, 0.f, 0.f, 0.f, 0.f, 0.f, 0.f, 0.f};

        if (wave < 3) {
            // Full 16-bin tile: compile-time trip count -> unrolled WMMA chain.
#pragma unroll
            for (int j = 0; j < 16; ++j) {
                const int kk = kk0 + j;
                const int s0 = (kk << 2) + (half << 1);
                // A 16x4 f32: lanes 0-15 -> K={4kk,4kk+1}, lanes 16-31 -> K={4kk+2,4kk+3}
                const v2f a = *(const v2f*)&Vs[m * NSAMP + s0];
                // P rows identical per k-step: 0.25 in column kk%16 (B stripes N=lane%16)
                const float bw = (m == j) ? 0.25f : 0.0f;
                const v2f bb = {bw, bw};
                acc = __builtin_amdgcn_wmma_f32_16x16x4_f32(
                    false, a, false, bb, (short)0, acc, false, false);
            }
        } else {
            // Tile 3: only bin 48 contributes (single k-step kk = 48).
            const int s0 = (48 << 2) + (half << 1);
            const v2f a = *(const v2f*)&Vs[m * NSAMP + s0];
            const float bw = (m == 0) ? 0.25f : 0.0f;
            const v2f bb = {bw, bw};
            acc = __builtin_amdgcn_wmma_f32_16x16x4_f32(
                false, a, false, bb, (short)0, acc, false, false);
        }

        const int bin = kk0 + m;               // D column -> global bin
        if (bin < NBINS) {
            float* op = out + (size_t)n * C_TOT + bin;
#pragma unroll
            for (int j = 0; j < 8; ++j) {      // D VGPR j -> M = j + 8*half
                op[(j + (half << 3)) * NBINS] = acc[j];
            }
        }
    }
}

extern "C" void kernel_launch(void* const* d_in, const int* in_sizes, int n_in,
                              void* d_out, int out_size, void* d_ws, size_t ws_size,
                              hipStream_t stream) {
    const float* feat = (const float*)d_in[0];   // [4, 784, 80, 80] f32
    const float* rois = (const float*)d_in[1];   // [512, 5] f32
    float* out = (float*)d_out;                  // [512, 16, 7, 7] f32
    const int nrois = in_sizes[1] / 5;
    psroi_align_wmma_kernel<<<nrois, 256, 0, stream>>>(feat, rois, out);
}